// RigidityLoss_30485677867094
// MI455X (gfx1250) — compile-verified
//
#include <hip/hip_runtime.h>
#include <hip/hip_bf16.h>

// Problem constants (match reference)
#define NPTS 32768
#define MPTS 16384
#define KNN  8
#define CDIM 16
#define LAMF 0.1f
#define EPSF 1e-6f

typedef float v2f __attribute__((ext_vector_type(2)));
typedef float v8f __attribute__((ext_vector_type(8)));

// ---------------------------------------------------------------------------
// Kernel 1: gather sampled points, precompute padded coords + squared norms
// ---------------------------------------------------------------------------
__global__ void __launch_bounds__(256)
rigidity_gather(const float* __restrict__ canon,   // [N,3]
                const float* __restrict__ trans,   // [N,3]
                const float* __restrict__ coeff,   // [N,16]
                const float* __restrict__ fdc,     // [N,1,3] flat
                const int*   __restrict__ indice,  // [M]
                float* __restrict__ tp4,           // [M,4] (x,y,z,0)
                float* __restrict__ sq,            // [M]
                float* __restrict__ tc,            // [M,16]
                float* __restrict__ tcol4)         // [M,4]
{
    int m = blockIdx.x * 256 + threadIdx.x;
    if (m >= MPTS) return;
    int i = indice[m];
    float x = canon[3*i+0] + trans[3*i+0];
    float y = canon[3*i+1] + trans[3*i+1];
    float z = canon[3*i+2] + trans[3*i+2];
    tp4[4*m+0] = x; tp4[4*m+1] = y; tp4[4*m+2] = z; tp4[4*m+3] = 0.0f;
    sq[m] = x*x + y*y + z*z;
#pragma unroll
    for (int c = 0; c < CDIM; ++c) tc[CDIM*m + c] = coeff[CDIM*i + c];
    tcol4[4*m+0] = fdc[3*i+0];
    tcol4[4*m+1] = fdc[3*i+1];
    tcol4[4*m+2] = fdc[3*i+2];
    tcol4[4*m+3] = 0.0f;
}

// ---------------------------------------------------------------------------
// Kernel 2: brute-force KNN via WMMA Gram tiles + fused per-row loss terms.
// One wave (32 threads) per block; each wave owns 32 query rows.
// D tile layout (ISA 7.12.2, 32-bit C/D 16x16): lane = column n (+16 split),
// vgpr v = row m (v, or v+8 for lanes 16..31).
// ---------------------------------------------------------------------------
__global__ void __launch_bounds__(32)
rigidity_knn_loss(const float* __restrict__ tp4,
                  const float* __restrict__ sq,
                  const float* __restrict__ tc,
                  const float* __restrict__ tcol4,
                  float* __restrict__ partial)     // [gridDim.x*2]
{
    __shared__ float tile[16][32];                 // [cand col][query row]

    const int lane = threadIdx.x;                  // 0..31 (wave32)
    const int nl   = lane & 15;
    const int hi   = lane >> 4;                    // K half: 0 -> K=0,1 ; 1 -> K=2,3
    const int r0   = blockIdx.x * 32;              // first query row of this wave

    const v2f* tp2 = (const v2f*)tp4;              // 2 x v2f per padded point

    // A fragments (16x4 f32, ISA layout): lane m = nl, K pair selected by hi.
    v2f a0 = tp2[(r0 +      nl) * 2 + hi];         // rows r0 .. r0+15
    v2f a1 = tp2[(r0 + 16 + nl) * 2 + hi];         // rows r0+16 .. r0+31

    float dist[KNN];
    int   idx[KNN];
#pragma unroll
    for (int k = 0; k < KNN; ++k) { dist[k] = 3.0e38f; idx[k] = 0; }

    const float sq_row = sq[r0 + lane];            // |query|^2 for the row this lane scans

    for (int c0 = 0; c0 < MPTS; c0 += 16) {
        // B fragment (4x16 f32): same per-lane pattern as A.
        v2f   b   = tp2[(c0 + nl) * 2 + hi];
        float sqn = sq[c0 + nl];

        v8f acc0 = {0.f,0.f,0.f,0.f,0.f,0.f,0.f,0.f};
        v8f acc1 = {0.f,0.f,0.f,0.f,0.f,0.f,0.f,0.f};
        acc0 = __builtin_amdgcn_wmma_f32_16x16x4_f32(false, a0, false, b,
                                                     (short)0, acc0, false, false);
        acc1 = __builtin_amdgcn_wmma_f32_16x16x4_f32(false, a1, false, b,
                                                     (short)0, acc1, false, false);

        // Fold |b|^2 - 2*dot here (lane owns a fixed candidate column nl),
        // stage transposed so the scanning lane reads its own row.
#pragma unroll
        for (int v = 0; v < 8; ++v) {
            tile[nl][     8*hi + v] = fmaf(-2.0f, acc0[v], sqn);
            tile[nl][16 + 8*hi + v] = fmaf(-2.0f, acc1[v], sqn);
        }
        __syncthreads();   // single-wave workgroup: effectively free

        // Lane `lane` scans row (r0+lane) of this 16-candidate tile.
#pragma unroll
        for (int j = 0; j < 16; ++j) {
            float d2 = sq_row + tile[j][lane];
            if (d2 < dist[KNN-1]) {                // usually false -> wave skips
                float dd = d2; int jj = c0 + j;
#pragma unroll
                for (int k = 0; k < KNN; ++k) {    // unrolled sorted insert
                    bool  lt = dd < dist[k];
                    float dn = lt ? dd      : dist[k];
                    float dxv= lt ? dist[k] : dd;
                    int   jn = lt ? jj      : idx[k];
                    int   jxv= lt ? idx[k]  : jj;
                    dist[k] = dn; idx[k] = jn; dd = dxv; jj = jxv;
                }
            }
        }
        __syncthreads();
    }

    // -------- per-row loss terms (row r = r0 + lane) --------
    const int r = r0 + lane;

    // 'surface' branch: || tp - mean(nn_points) + eps ||
    float mx = 0.f, my = 0.f, mz = 0.f;
#pragma unroll
    for (int k = 0; k < KNN; ++k) {
        int j = idx[k];
        mx += tp4[4*j+0]; my += tp4[4*j+1]; mz += tp4[4*j+2];
    }
    mx *= 0.125f; my *= 0.125f; mz *= 0.125f;
    float px = tp4[4*r+0], py = tp4[4*r+1], pz = tp4[4*r+2];
    float dxe = px - mx + EPSF, dye = py - my + EPSF, dze = pz - mz + EPSF;
    float s1 = sqrtf(dxe*dxe + dye*dye + dze*dze);

    // 'coeff' branch
    float cr0 = tcol4[4*r+0], cr1 = tcol4[4*r+1], cr2 = tcol4[4*r+2];
    float tcr[CDIM];
#pragma unroll
    for (int c = 0; c < CDIM; ++c) tcr[c] = tc[CDIM*r + c];

    float s2 = 0.f;
#pragma unroll
    for (int k = 0; k < KNN; ++k) {
        int j = idx[k];
        float e0 = cr0 - tcol4[4*j+0] + EPSF;
        float e1 = cr1 - tcol4[4*j+1] + EPSF;
        float e2 = cr2 - tcol4[4*j+2] + EPSF;
        float cd = sqrtf(e0*e0 + e1*e1 + e2*e2);
        float sv = 0.f;
#pragma unroll
        for (int c = 0; c < CDIM; ++c) {
            float f = tcr[c] - tc[CDIM*j + c] + EPSF;
            sv += f * f;
        }
        sv = sqrtf(sv);
        float dw = expf(-LAMF * dist[k] * dist[k]);   // deform_dists are squared d2
        float cw = expf(-LAMF * cd * cd);
        s2 += cw * dw * sv;
    }

    // wave32 reduction, fixed order -> deterministic
#pragma unroll
    for (int off = 16; off; off >>= 1) {
        s1 += __shfl_xor(s1, off);
        s2 += __shfl_xor(s2, off);
    }
    if (lane == 0) {
        partial[2*blockIdx.x + 0] = s1;
        partial[2*blockIdx.x + 1] = s2;
    }
}

// ---------------------------------------------------------------------------
// Kernel 3: deterministic final reduction of per-block partials
// ---------------------------------------------------------------------------
__global__ void __launch_bounds__(32)
rigidity_finalize(const float* __restrict__ partial, float* __restrict__ out)
{
    const int nblk = MPTS / 32;
    int t = threadIdx.x;
    float s1 = 0.f, s2 = 0.f;
    for (int b = t; b < nblk; b += 32) {
        s1 += partial[2*b + 0];
        s2 += partial[2*b + 1];
    }
#pragma unroll
    for (int off = 16; off; off >>= 1) {
        s1 += __shfl_xor(s1, off);
        s2 += __shfl_xor(s2, off);
    }
    if (t == 0)
        out[0] = s1 / (float)MPTS + s2 / ((float)MPTS * (float)KNN);
}

// ---------------------------------------------------------------------------
extern "C" void kernel_launch(void* const* d_in, const int* in_sizes, int n_in,
                              void* d_out, int out_size, void* d_ws, size_t ws_size,
                              hipStream_t stream)
{
    const float* canon  = (const float*)d_in[0];   // canon_xyz       [N,3]
    const float* trans  = (const float*)d_in[1];   // pred_translation[N,3]
    const float* coeff  = (const float*)d_in[2];   // motion_coeff    [N,16]
    const float* fdc    = (const float*)d_in[3];   // features_dc     [N,1,3]
    const int*   indice = (const int*)d_in[4];     // indice          [M]

    float* ws      = (float*)d_ws;
    float* tp4     = ws;                       // M*4
    float* sq      = tp4   + (size_t)MPTS * 4; // M
    float* tc      = sq    + (size_t)MPTS;     // M*16
    float* tcol4   = tc    + (size_t)MPTS * CDIM; // M*4
    float* partial = tcol4 + (size_t)MPTS * 4; // (M/32)*2

    rigidity_gather<<<MPTS / 256, 256, 0, stream>>>(canon, trans, coeff, fdc,
                                                    indice, tp4, sq, tc, tcol4);
    rigidity_knn_loss<<<MPTS / 32, 32, 0, stream>>>(tp4, sq, tc, tcol4, partial);
    rigidity_finalize<<<1, 32, 0, stream>>>(partial, (float*)d_out);
}